// PermutohedralLayer_90305982365926
// MI455X (gfx1250) — compile-verified
//
#include <hip/hip_runtime.h>

typedef __attribute__((ext_vector_type(2))) float v2f;
typedef __attribute__((ext_vector_type(8))) float v8f;

#define HW_H   96
#define HW_W   96
#define NPIX   (HW_H * HW_W)   // 9216
#define NB     2
#define NC     21
#define NT     (NPIX / 16)     // 576 j/i tiles
#define NWAVE  8               // waves per block (wave32)
#define KSTR   18              // padded LDS row stride (floats) for K tile: 8B-aligned, bank-skewed

static __device__ __forceinline__ v8f wmma_f32(v2f a, v2f b, v8f c) {
    // V_WMMA_F32_16X16X4_F32 : D = A(16x4) * B(4x16) + C(16x16), fp32
    return __builtin_amdgcn_wmma_f32_16x16x4_f32(false, a, false, b, (short)0, c, false, false);
}

// ---------------------------------------------------------------------------
// Prep A: feats[b][n][8] = { y/8, x/8, img0*2, img1*2, img2*2, 0,0,0 }, sq[b][n]
// ---------------------------------------------------------------------------
__global__ void __launch_bounds__(256) pgf_prep_feats(const float* __restrict__ img,
                                                      float* __restrict__ feats,
                                                      float* __restrict__ sq) {
    int idx = blockIdx.x * blockDim.x + threadIdx.x;
    if (idx >= NB * NPIX) return;
    int b = idx / NPIX;
    int n = idx - b * NPIX;
    float f[8];
    f[0] = (float)(n / HW_W) * 0.125f;   // y / THETA_ALPHA
    f[1] = (float)(n % HW_W) * 0.125f;   // x / THETA_ALPHA
    const float* ip = img + (size_t)b * 3 * NPIX + n;
    f[2] = ip[0]        * 2.0f;          // / THETA_BETA
    f[3] = ip[NPIX]     * 2.0f;
    f[4] = ip[2 * NPIX] * 2.0f;
    f[5] = 0.0f; f[6] = 0.0f; f[7] = 0.0f;
    float s = 0.0f;
#pragma unroll
    for (int v = 0; v < 8; ++v) s += f[v] * f[v];
    float* fp = feats + (size_t)idx * 8;
#pragma unroll
    for (int v = 0; v < 8; ++v) fp[v] = f[v];
    sq[idx] = s;
}

// ---------------------------------------------------------------------------
// Prep B: VT[b][jt][c(0..31)][jj(0..15)] = vals[b][c][jt*16+jj], zero-padded c
// ---------------------------------------------------------------------------
__global__ void __launch_bounds__(256) pgf_prep_vt(const float* __restrict__ vals,
                                                   float* __restrict__ vt) {
    int idx = blockIdx.x * blockDim.x + threadIdx.x;
    if (idx >= NB * NT * 32) return;
    int c  = idx & 31;
    int jt = (idx >> 5) % NT;
    int b  = idx / (32 * NT);
    float* dst = vt + (size_t)idx * 16;
    if (c < NC) {
        const float* src = vals + ((size_t)b * NC + c) * NPIX + jt * 16;
#pragma unroll
        for (int jj = 0; jj < 16; ++jj) dst[jj] = src[jj];
    } else {
#pragma unroll
        for (int jj = 0; jj < 16; ++jj) dst[jj] = 0.0f;
    }
}

// ---------------------------------------------------------------------------
// Main: block = (b, i-tile). 8 waves split the j-tiles; LDS reduce at the end.
// ---------------------------------------------------------------------------
__global__ void __launch_bounds__(256) pgf_main(const float* __restrict__ feats,
                                                const float* __restrict__ sq,
                                                const float* __restrict__ vt,
                                                float* __restrict__ out) {
    __shared__ float lds_k[NWAVE * 16 * KSTR];   // per-wave K-tile scratch
    __shared__ float lds_r[NWAVE * 512];         // cross-wave reduction

    const int tid = threadIdx.x;
    const int wv  = tid >> 5;
    const int l   = tid & 31;
    const int col = l & 15;    // N index of fragments / j column / c column
    const int p   = l >> 4;    // lane half
    const int it     = blockIdx.x;
    const int b      = blockIdx.y;
    const int i_base = it * 16;
    const size_t fbase = (size_t)b * NPIX;

    // A-fragment of i-tile features (constant over the j loop).
    // fp32 16x4 A layout: lane(l) holds F[l%16][2*(l/16) + {0,1}] per 2-VGPR frag.
    const float* fi = feats + (fbase + i_base + col) * 8;
    v2f alo = *(const v2f*)(fi + 2 * p);
    v2f ahi = *(const v2f*)(fi + 4 + 2 * p);

    // |f_i|^2 for the 8 rows this lane accumulates in C/D layout (M = v + 8p)
    float si[8];
    const float* sqi = sq + fbase + i_base + 8 * p;
#pragma unroll
    for (int v = 0; v < 8; ++v) si[v] = sqi[v];

    v8f acc0 = {};  // c = 0..15
    v8f acc1 = {};  // c = 16..31
    float* kt = lds_k + wv * 16 * KSTR;

    for (int jt = wv; jt < NT; jt += NWAVE) {
        const int j_base = jt * 16;

        // B-fragment of j-tile features: same per-lane load pattern as A.
        const float* fj = feats + (fbase + j_base + col) * 8;
        v2f blo = *(const v2f*)(fj + 2 * p);
        v2f bhi = *(const v2f*)(fj + 4 + 2 * p);
        float sj = sq[fbase + j_base + col];

        // Stage 1: G = F_i * F_j^T  (K-dim 8 = two K=4 WMMAs)
        v8f g = {};
        g = wmma_f32(alo, blo, g);
        g = wmma_f32(ahi, bhi, g);

        // K = exp(-0.5*max(d2,0)); write 16x16 tile to LDS in row-major (C/D layout out)
#pragma unroll
        for (int v = 0; v < 8; ++v) {
            float d2 = si[v] + sj - 2.0f * g[v];
            float kv = __expf(-0.5f * fmaxf(d2, 0.0f));
            kt[(v + 8 * p) * KSTR + col] = kv;
        }

        // Prefetch next j-tile features for this wave
        if (jt + NWAVE < NT)
            __builtin_prefetch(feats + (fbase + j_base + 16 * NWAVE + col) * 8, 0, 3);

        // Stage 2: acc += K_tile(16x16) * V(16x32), four K=4 chunks, two c-tiles.
        const float* vtile = vt + (size_t)(b * NT + jt) * 32 * 16;
#pragma unroll
        for (int kk = 0; kk < 4; ++kk) {
            v2f ka  = *(const v2f*)(kt + col * KSTR + 4 * kk + 2 * p);          // A-frag of K
            v2f vb0 = *(const v2f*)(vtile + (col     ) * 16 + 4 * kk + 2 * p);  // B-frag c 0..15
            v2f vb1 = *(const v2f*)(vtile + (col + 16) * 16 + 4 * kk + 2 * p);  // B-frag c 16..31
            acc0 = wmma_f32(ka, vb0, acc0);
            acc1 = wmma_f32(ka, vb1, acc1);
        }
    }

    // Cross-wave reduction via LDS
#pragma unroll
    for (int v = 0; v < 8; ++v) {
        int row = v + 8 * p;
        lds_r[wv * 512 + row * 32 + col]      = acc0[v];
        lds_r[wv * 512 + row * 32 + col + 16] = acc1[v];
    }
    __syncthreads();
    for (int e = tid; e < 512; e += 256) {
        float s = 0.0f;
#pragma unroll
        for (int w2 = 0; w2 < NWAVE; ++w2) s += lds_r[w2 * 512 + e];
        int row = e >> 5;
        int c   = e & 31;
        if (c < NC) out[((size_t)b * NC + c) * NPIX + i_base + row] = s;
    }
}

// ---------------------------------------------------------------------------
extern "C" void kernel_launch(void* const* d_in, const int* in_sizes, int n_in,
                              void* d_out, int out_size, void* d_ws, size_t ws_size,
                              hipStream_t stream) {
    const float* cur_state   = (const float*)d_in[0];  // [2,21,96,96]
    const float* input_image = (const float*)d_in[1];  // [2,3,96,96]
    float* out = (float*)d_out;                        // [2,21,96,96]

    float* ws    = (float*)d_ws;
    float* feats = ws;                                  // NB*NPIX*8
    float* sq    = feats + (size_t)NB * NPIX * 8;       // NB*NPIX
    float* vtws  = sq    + (size_t)NB * NPIX;           // NB*NT*32*16
    (void)in_sizes; (void)n_in; (void)out_size; (void)ws_size;

    {
        int nthr = NB * NPIX;
        pgf_prep_feats<<<(nthr + 255) / 256, 256, 0, stream>>>(input_image, feats, sq);
    }
    {
        int nthr = NB * NT * 32;
        pgf_prep_vt<<<(nthr + 255) / 256, 256, 0, stream>>>(cur_state, vtws);
    }
    {
        dim3 grid(NT, NB);
        pgf_main<<<grid, 256, 0, stream>>>(feats, sq, vtws, out);
    }
}